// SSMLayer_5282809774601
// MI455X (gfx1250) — compile-verified
//
#include <hip/hip_runtime.h>
#include <hip/hip_bf16.h>

// ---------------------------------------------------------------------------
// S4 SSM layer on MI455X (gfx1250), FP32 WMMA chunked-scan formulation.
//
//   x_t = Ab x_{t-1} + Bb u_t ,  y_t = C x_t + D u_t ,  x_{-1} = 0
// Chunk T=64, N=64, L=8192, BATCH=2048:
//   y_chunk = Mconv @ U + P @ x_in + D*U      (Mconv[i][j]=K[i-j], P[i]=C Ab^{i+1})
//   x_out   = AbT @ x_in + W @ U              (AbT=Ab^64, W[:,j]=Ab^{63-j} Bb)
// Four 64x64x16 GEMMs per chunk on V_WMMA_F32_16X16X4_F32; next chunk's U is
// DMA'd into an LDS double buffer with GLOBAL_LOAD_ASYNC_TO_LDS_B128 while
// the current chunk computes (per-wave s_wait_asynccnt + WG barrier handoff).
// ---------------------------------------------------------------------------

typedef __attribute__((ext_vector_type(2))) float v2f;
typedef __attribute__((ext_vector_type(8))) float v8f;

#define NST   64      // state dim N
#define TCH   64      // chunk length T
#define SEQL  8192
#define NCHUNK (SEQL / TCH)     // 128
#define NBATCH 2048
#define GROUP  16               // batch cols per workgroup (WMMA N)
#define NGROUP (NBATCH / GROUP) // 128
#define ROWP  68                // padded LDS row: 16B-aligned rows, bank-safe

// workspace float offsets
#define MC_OFF  0               // Mconv 64x64
#define P_OFF   4096            // P     64x64
#define W_OFF   8192            // W     64x64
#define ABT_OFF 12288           // Ab^64 64x64

static __device__ inline v8f wmma4(v2f a, v2f b, v8f c) {
  // 8-arg form: (neg_a, A, neg_b, B, c_mod, C, reuse_a, reuse_b)
  return __builtin_amdgcn_wmma_f32_16x16x4_f32(false, a, false, b, (short)0, c,
                                               false, false);
}

// Async DMA: 16 contiguous bytes global -> LDS per lane, tracked by ASYNCcnt.
static __device__ inline void async_copy_b128(unsigned lds_addr,
                                              const void* gptr) {
  asm volatile("global_load_async_to_lds_b128 %0, %1, off"
               :: "v"(lds_addr), "v"(gptr)
               : "memory");
}

// 64x64 @ 64x64 -> 64x64 in LDS, 256 threads, scalar (setup only)
static __device__ void mm64(const float* Sa, int lda, const float* Sb, int ldb,
                            float* Dst, int ldd, int tid) {
  __syncthreads();
  for (int idx = tid; idx < 64 * 64; idx += 256) {
    int i = idx >> 6, j = idx & 63;
    float acc = 0.f;
    for (int k = 0; k < 64; ++k) acc += Sa[i * lda + k] * Sb[k * ldb + j];
    Dst[i * ldd + j] = acc;
  }
  __syncthreads();
}

// ---------------------------------------------------------------------------
// Kernel 1: discretization + constant-matrix factory (single workgroup).
// ---------------------------------------------------------------------------
__global__ __launch_bounds__(256) void ssm_setup(const float* __restrict__ A,
                                                 const float* __restrict__ Bv,
                                                 const float* __restrict__ Cv,
                                                 const float* __restrict__ logstep,
                                                 float* __restrict__ ws) {
  __shared__ float aug[64][130];  // [ I - s/2 A | I ] -> after GJ: [ * | BL ]
  __shared__ float M1[64][64];    // I + s/2 A, later squaring temp
  __shared__ float vK[64], Kt[64], rP[64], rP2[64];
  __shared__ float stepsh;

  const int tid = threadIdx.x;
  if (tid == 0) stepsh = __expf(logstep[0]);
  __syncthreads();
  const float s2 = 0.5f * stepsh;

  for (int idx = tid; idx < 64 * 64; idx += 256) {
    int i = idx >> 6, j = idx & 63;
    float a = A[idx];
    float d = (i == j) ? 1.0f : 0.0f;
    aug[i][j]      = d - s2 * a;
    aug[i][64 + j] = d;
    M1[i][j]       = d + s2 * a;
  }
  __syncthreads();

  // Gauss-Jordan (pivot-free: I - (s/2)A with s <= 0.1 is diag dominant)
  const int ri = tid >> 2;          // row handled in elimination
  const int jc = (tid & 3) << 5;    // 32-column slab
  for (int k = 0; k < 64; ++k) {
    float invp = 1.0f / aug[k][k];
    __syncthreads();
    if (tid < 128) aug[k][tid] *= invp;
    __syncthreads();
    float f = aug[ri][k];
    __syncthreads();
    if (ri != k) {
      for (int j = jc; j < jc + 32; ++j) aug[ri][j] -= f * aug[k][j];
    }
    __syncthreads();
  }

  // Ab = BL @ (I + s/2 A) written into left half of aug (identity there now)
  for (int idx = tid; idx < 64 * 64; idx += 256) {
    int i = idx >> 6, j = idx & 63;
    float acc = 0.f;
    for (int k = 0; k < 64; ++k) acc += aug[i][64 + k] * M1[k][j];
    aug[i][j] = acc;
  }
  __syncthreads();

  // Bb = step * BL @ B   (seed for the K / W scan)
  if (tid < 64) {
    float acc = 0.f;
    for (int k = 0; k < 64; ++k) acc += aug[tid][64 + k] * Bv[k];
    vK[tid] = stepsh * acc;
  }
  __syncthreads();

  // scan v_l = Ab^l Bb :  K[l] = C.v_l ,  W[:,63-l] = v_l
  for (int l = 0; l < 64; ++l) {
    if (tid < 64) ws[W_OFF + tid * 64 + (63 - l)] = vK[tid];
    if (tid == 0) {
      float acc = 0.f;
      for (int k = 0; k < 64; ++k) acc += Cv[k] * vK[k];
      Kt[l] = acc;
    }
    __syncthreads();
    float nv = 0.f;
    if (tid < 64) {
      float acc = 0.f;
      for (int k = 0; k < 64; ++k) acc += aug[tid][k] * vK[k];
      nv = acc;
    }
    __syncthreads();
    if (tid < 64) vK[tid] = nv;
    __syncthreads();
  }

  // scan r_{i+1} = r_i Ab starting at C :  P[i][:] = C Ab^{i+1}
  if (tid < 64) rP[tid] = Cv[tid];
  __syncthreads();
  for (int i = 0; i < 64; ++i) {
    if (tid < 64) {
      float acc = 0.f;
      for (int k = 0; k < 64; ++k) acc += rP[k] * aug[k][tid];
      rP2[tid] = acc;
    }
    __syncthreads();
    if (tid < 64) {
      rP[tid] = rP2[tid];
      ws[P_OFF + i * 64 + tid] = rP2[tid];
    }
    __syncthreads();
  }

  // AbT = Ab^64 by 6 squarings (BL in aug-right is dead now; M1 is dead)
  float* augL = &aug[0][0];
  float* augR = &aug[0][64];
  float* m1   = &M1[0][0];
  mm64(augL, 130, augL, 130, m1, 64, tid);    // ^2
  mm64(m1, 64, m1, 64, augR, 130, tid);       // ^4
  mm64(augR, 130, augR, 130, m1, 64, tid);    // ^8
  mm64(m1, 64, m1, 64, augR, 130, tid);       // ^16
  mm64(augR, 130, augR, 130, m1, 64, tid);    // ^32
  mm64(m1, 64, m1, 64, augR, 130, tid);       // ^64
  for (int idx = tid; idx < 64 * 64; idx += 256)
    ws[ABT_OFF + idx] = aug[idx >> 6][64 + (idx & 63)];

  // Mconv[i][j] = K[i-j] (lower-triangular Toeplitz)
  for (int idx = tid; idx < 64 * 64; idx += 256) {
    int i = idx >> 6, j = idx & 63;
    ws[MC_OFF + idx] = (j <= i) ? Kt[i - j] : 0.0f;
  }
}

// ---------------------------------------------------------------------------
// Kernel 2: fused chunked scan. 128 WGs x 128 threads (4 waves).
// Wave w computes row band [16w,16w+16) of all 64-row GEMM outputs.
//
// A-fragment layout (32-bit A 16x4, ISA 7.12.2): lanes 0-15 hold K={0,1},
// lanes 16-31 hold K={2,3}, row M = lane&15, across the 2 VGPRs.
// B-fragment layout (4x16) assumed mirrored: lane -> column N = lane&15,
// K pair {0,1} for lanes 0-15, {2,3} for lanes 16-31.
//
// LDS staging is batch-major with rows padded to 68 floats: rows stay
// 16B-aligned (async B128 stores legal), B-fragment ds_load_b64 reads hit
// banks (4n+kb) mod 64 -> conflict-free, same for accumulator writebacks.
// ---------------------------------------------------------------------------
__global__ __launch_bounds__(128) void ssm_main(const float* __restrict__ u,
                                                const float* __restrict__ Dp,
                                                const float* __restrict__ ws,
                                                float* __restrict__ y) {
  __shared__ __align__(16) float Uc[2][GROUP][ROWP];  // double-buffered U chunk
  __shared__ __align__(16) float Xs[GROUP][ROWP];     // state  [batch][state]
  __shared__ __align__(16) float Yl[GROUP][ROWP];     // output [batch][time]

  const int tid  = threadIdx.x;
  const int w    = tid >> 5;        // wave id 0..3 -> M band
  const int lane = tid & 31;
  const int b0   = blockIdx.x * GROUP;
  const float Dval = Dp[0];

  // Preload constant A-fragments for this wave's 16-row band: 16 K-tiles each.
  v2f fM[16], fP[16], fW[16], fA[16];
  {
    const int row = 16 * w + (lane & 15);
    const int kh  = (lane >> 4) * 2;  // 0 or 2
    const float* Mc = ws + MC_OFF + row * 64;
    const float* Pm = ws + P_OFF + row * 64;
    const float* Wm = ws + W_OFF + row * 64;
    const float* Am = ws + ABT_OFF + row * 64;
#pragma unroll
    for (int kt = 0; kt < 16; ++kt) {
      int kb = kt * 4 + kh;
      v2f t;
      t.x = Mc[kb]; t.y = Mc[kb + 1]; fM[kt] = t;
      t.x = Pm[kb]; t.y = Pm[kb + 1]; fP[kt] = t;
      t.x = Wm[kb]; t.y = Wm[kb + 1]; fW[kt] = t;
      t.x = Am[kb]; t.y = Am[kb + 1]; fA[kt] = t;
    }
  }

  // x_{-1} = 0
  for (int idx = tid; idx < GROUP * ROWP; idx += 128) (&Xs[0][0])[idx] = 0.f;

  // Kick off async DMA of chunk 0 into buffer 0 (2 B128 instrs per thread:
  // t2 -> batch row r = t2>>4, 16B slice q = t2&15 of that row).
#pragma unroll
  for (int it = 0; it < 2; ++it) {
    int t2 = tid + it * 128;
    int r = t2 >> 4, q = t2 & 15;
    async_copy_b128((unsigned)(uintptr_t)&Uc[0][r][q * 4],
                    u + (size_t)(b0 + r) * SEQL + q * 4);
  }

  const int n   = lane & 15;        // WMMA N index (batch col in group)
  const int khb = (lane >> 4) * 2;  // K sub-pair base

  for (int c = 0; c < NCHUNK; ++c) {
    const int buf = c & 1;

    // Issue next chunk's DMA into the other buffer, then drain this chunk's
    // copies (own wave) and barrier (publishes all waves' LDS writes).
    if (c + 1 < NCHUNK) {
#pragma unroll
      for (int it = 0; it < 2; ++it) {
        int t2 = tid + it * 128;
        int r = t2 >> 4, q = t2 & 15;
        async_copy_b128((unsigned)(uintptr_t)&Uc[buf ^ 1][r][q * 4],
                        u + (size_t)(b0 + r) * SEQL + (size_t)(c + 1) * TCH +
                            q * 4);
      }
      asm volatile("s_wait_asynccnt 0x2" ::: "memory");
    } else {
      asm volatile("s_wait_asynccnt 0x0" ::: "memory");
    }
    __syncthreads();

    v8f acc  = {};  // Mconv@U + P@x   (this wave's 16 output rows)
    v8f xacc = {};  // W@U + AbT@x     (this wave's 16 state rows)
#pragma unroll
    for (int kt = 0; kt < 16; ++kt) {  // K over time within chunk
      int kb = kt * 4 + khb;
      v2f b;
      b.x = Uc[buf][n][kb];
      b.y = Uc[buf][n][kb + 1];
      acc  = wmma4(fM[kt], b, acc);
      xacc = wmma4(fW[kt], b, xacc);
    }
#pragma unroll
    for (int kt = 0; kt < 16; ++kt) {  // K over state dim
      int kb = kt * 4 + khb;
      v2f b;
      b.x = Xs[n][kb];
      b.y = Xs[n][kb + 1];
      acc  = wmma4(fP[kt], b, acc);
      xacc = wmma4(fA[kt], b, xacc);
    }
    __syncthreads();  // all waves done reading old Xs

    // C/D layout: VGPR r -> M = r + 8*(lane>=16), N = lane&15
#pragma unroll
    for (int r = 0; r < 8; ++r) {
      int gi = 16 * w + r + (lane >> 4) * 8;
      Xs[n][gi] = xacc[r];
      Yl[n][gi] = acc[r] + Dval * Uc[buf][n][gi];
    }
    __syncthreads();

    // Coalesced float4 store of the chunk's outputs.
    for (int t2 = tid; t2 < 256; t2 += 128) {
      int r = t2 >> 4, c4 = t2 & 15;
      const float4 val = *(const float4*)&Yl[r][c4 * 4];
      *(float4*)(y + (size_t)(b0 + r) * SEQL + c * TCH + c4 * 4) = val;
    }
    __syncthreads();
  }
}

extern "C" void kernel_launch(void* const* d_in, const int* in_sizes, int n_in,
                              void* d_out, int out_size, void* d_ws,
                              size_t ws_size, hipStream_t stream) {
  const float* u  = (const float*)d_in[0];
  const float* A  = (const float*)d_in[1];
  const float* Bv = (const float*)d_in[2];
  const float* Cv = (const float*)d_in[3];
  const float* Dp = (const float*)d_in[4];
  const float* ls = (const float*)d_in[5];
  float* ws = (float*)d_ws;
  float* y  = (float*)d_out;

  ssm_setup<<<1, 256, 0, stream>>>(A, Bv, Cv, ls, ws);
  ssm_main<<<NGROUP, 128, 0, stream>>>(u, Dp, ws, y);
}